// Manager_57380763075089
// MI455X (gfx1250) — compile-verified
//
#include <hip/hip_runtime.h>
#include <hip/hip_bf16.h>
#include <math.h>
#include <stdint.h>

// -------- CDNA5 fp32 WMMA helpers (V_WMMA_F32_16X16X4_F32) --------
typedef float v2f __attribute__((ext_vector_type(2)));
typedef float v8f __attribute__((ext_vector_type(8)));

static __device__ inline v8f vzero8() {
    v8f z;
#pragma unroll
    for (int i = 0; i < 8; ++i) z[i] = 0.0f;
    return z;
}

static __device__ inline v8f wmma4(v2f a, v2f b, v8f c) {
    // 8 args: (neg_a, A, neg_b, B, c_mod, C, reuse_a, reuse_b)
    return __builtin_amdgcn_wmma_f32_16x16x4_f32(false, a, false, b,
                                                 (short)0, c, false, false);
}

// =====================================================================
// Kernel 1: h = features[2048x2048] @ W_arch[2048x1000] + b_arch
// Block = 256 threads = 8 waves. Wave (wr,wc) computes a 16x64 strip.
// Block tile = 64 rows x 128 cols. Grid = (32, 8).
// Column guard handled by ADDRESS CLAMPING (no exec divergence): columns
// n >= 1000 compute garbage that is simply never stored.
// =====================================================================
__global__ void k_gemm_arch(const float* __restrict__ feat,
                            const float* __restrict__ W,
                            const float* __restrict__ bias,
                            float* __restrict__ h) {
    const int lane  = threadIdx.x & 31;
    const int wave  = threadIdx.x >> 5;     // 0..7
    const int wr    = wave >> 1;            // 0..3  (row group)
    const int wc    = wave & 1;             // 0..1  (col group)
    const int m16   = lane & 15;
    const int koff  = (lane < 16) ? 0 : 2;
    const int rhalf = (lane < 16) ? 0 : 8;

    const int rowA    = blockIdx.x * 64 + wr * 16 + m16;
    const int colbase = blockIdx.y * 128 + wc * 64;

    // clamped per-lane column indices (loop-invariant, no branches in loop)
    const int n0 = min(colbase +  0 + m16, 999);
    const int n1 = min(colbase + 16 + m16, 999);
    const int n2 = min(colbase + 32 + m16, 999);
    const int n3 = min(colbase + 48 + m16, 999);

    v8f c0 = vzero8(), c1 = vzero8(), c2 = vzero8(), c3 = vzero8();

    const float* __restrict__ pa = feat + (size_t)rowA * 2048 + koff;
    const float* __restrict__ p0 = W + (size_t)koff * 1000 + n0;
    const float* __restrict__ p1 = W + (size_t)koff * 1000 + n1;
    const float* __restrict__ p2 = W + (size_t)koff * 1000 + n2;
    const float* __restrict__ p3 = W + (size_t)koff * 1000 + n3;

#pragma unroll 2
    for (int kk = 0; kk < 2048; kk += 4) {
        v2f a = *(const v2f*)pa;            // A[m, kr], A[m, kr+1]
        v2f b0; b0.x = p0[0]; b0.y = p0[1000];
        v2f b1; b1.x = p1[0]; b1.y = p1[1000];
        v2f b2; b2.x = p2[0]; b2.y = p2[1000];
        v2f b3; b3.x = p3[0]; b3.y = p3[1000];
        c0 = wmma4(a, b0, c0);
        c1 = wmma4(a, b1, c1);
        c2 = wmma4(a, b2, c2);
        c3 = wmma4(a, b3, c3);
        pa += 4; p0 += 4000; p1 += 4000; p2 += 4000; p3 += 4000;
    }

    const int rowD = blockIdx.x * 64 + wr * 16 + rhalf;
    v8f accs[4] = {c0, c1, c2, c3};
#pragma unroll
    for (int j = 0; j < 4; ++j) {
        const int n = colbase + 16 * j + m16;
        if (n < 1000) {
            const float bb = bias[n];
#pragma unroll
            for (int r = 0; r < 8; ++r) {
                h[(size_t)(rowD + r) * 1000 + n] = accs[j][r] + bb;
            }
        }
    }
}

// =====================================================================
// Kernel 2: row-wise L2 normalize h (2048 rows x 1000 cols), in place.
// =====================================================================
__global__ void k_rownorm(float* __restrict__ h) {
    __shared__ float red[256];
    const int row = blockIdx.x;
    float ss = 0.0f;
    for (int i = threadIdx.x; i < 1000; i += 256) {
        const float v = h[(size_t)row * 1000 + i];
        ss += v * v;
    }
    red[threadIdx.x] = ss;
    __syncthreads();
    for (int s = 128; s > 0; s >>= 1) {
        if (threadIdx.x < s) red[threadIdx.x] += red[threadIdx.x + s];
        __syncthreads();
    }
    const float scale = 1.0f / fmaxf(sqrtf(red[0]), 1e-12f);
    for (int i = threadIdx.x; i < 1000; i += 256) {
        h[(size_t)row * 1000 + i] *= scale;
    }
}

// =====================================================================
// Kernel 3: fused per-task QKV attention + emb L2 norm + projection.
// Block = 512 threads = 16 waves = 16 heads; 16 rows of h per block.
// Grid = 128 blocks.
//  - h tile (16x1000) staged to LDS once via global_load_async_to_lds_b128
//    (ASYNCcnt path), LDS row pitch 1002 for conflict-free A-fragment reads.
//  - per t: wave=head computes xq/xk/xv [16x64] via 12 WMMA / k-step,
//    scores via shfl reduce -> LDS, softmax over heads by wave 0,
//    att*xv accumulated in registers across (t,head).
//  - final: ds_add_f32 combine, L2 norm, 16x100 projection via WMMA.
// =====================================================================
#define HP 1002  // padded LDS row pitch (floats)

__global__ void k_attn(const float* __restrict__ h,
                       const float* __restrict__ Q,
                       const float* __restrict__ K,
                       const float* __restrict__ V,
                       const float* __restrict__ Wp,
                       const float* __restrict__ bp,
                       float* __restrict__ out) {
    __shared__ float hs[16 * HP];         // staged h tile, padded pitch
    __shared__ float scores_s[16 * 16];   // [row][head]
    __shared__ float att_s[16 * 16];      // [row][head]
    __shared__ float emb_s[16 * 64];      // [row][e]
    __shared__ float invn_s[16];

    const int lane  = threadIdx.x & 31;
    const int wave  = threadIdx.x >> 5;   // 0..15 == head index
    const int m16   = lane & 15;
    const int koff  = (lane < 16) ? 0 : 2;
    const int rhalf = (lane < 16) ? 0 : 8;
    const int rowbase = blockIdx.x * 16;

    // ---- stage h tile: 16 rows x 1000 floats = 4000 x 16-byte chunks ----
    for (int c = threadIdx.x; c < 4000; c += 512) {
        const int row = c / 250;            // 250 chunks per row
        const int col = (c - row * 250) * 4;
        unsigned ldsoff = (unsigned)(uintptr_t)(&hs[row * HP + col]);
        unsigned long long ga =
            (unsigned long long)(uintptr_t)(h + (size_t)(rowbase + row) * 1000 + col);
        asm volatile("global_load_async_to_lds_b128 %0, %1, off"
                     :: "v"(ldsoff), "v"(ga) : "memory");
    }
    for (int i = threadIdx.x; i < 16 * 64; i += 512) emb_s[i] = 0.0f;
    asm volatile("s_wait_asynccnt 0x0" ::: "memory");
    __syncthreads();

    // persistent emb accumulator: element (m=r+rhalf, e=16j+m16)
    v8f eacc[4];
#pragma unroll
    for (int j = 0; j < 4; ++j) eacc[j] = vzero8();

    const float* __restrict__ arow = hs + m16 * HP + koff;  // LDS A fragments

    for (int t = 0; t < 20; ++t) {
        const size_t base = (size_t)(t * 16 + wave) * 64000;  // 1000*64
        const float* __restrict__ qrow = Q + base + koff * 64 + m16;
        const float* __restrict__ krow = K + base + koff * 64 + m16;
        const float* __restrict__ vrow = V + base + koff * 64 + m16;
        if (t < 19) {  // hint next head's slice (next t for wave 15) toward L2
            __builtin_prefetch(Q + base + 64000, 0, 1);
            __builtin_prefetch(K + base + 64000, 0, 1);
            __builtin_prefetch(V + base + 64000, 0, 1);
        }

        v8f cq[4], ck[4], cv[4];
#pragma unroll
        for (int j = 0; j < 4; ++j) { cq[j] = vzero8(); ck[j] = vzero8(); cv[j] = vzero8(); }

#pragma unroll 2
        for (int kk = 0; kk < 1000; kk += 4) {
            v2f a = *(const v2f*)(arow + kk);   // ds_load_b64
#pragma unroll
            for (int j = 0; j < 4; ++j) {
                const int o = 16 * j;           // immediate offsets off running ptrs
                v2f bq; bq.x = qrow[o]; bq.y = qrow[o + 64];
                v2f bk; bk.x = krow[o]; bk.y = krow[o + 64];
                v2f bv; bv.x = vrow[o]; bv.y = vrow[o + 64];
                cq[j] = wmma4(a, bq, cq[j]);
                ck[j] = wmma4(a, bk, ck[j]);
                cv[j] = wmma4(a, bv, cv[j]);
            }
            qrow += 256; krow += 256; vrow += 256;
        }

        // scores[m] = (1/64) * sum_e xq*xk ; butterfly-reduce per 16-lane half
        float part[8];
#pragma unroll
        for (int r = 0; r < 8; ++r) {
            float p = 0.0f;
#pragma unroll
            for (int j = 0; j < 4; ++j) p += cq[j][r] * ck[j][r];
#pragma unroll
            for (int mask = 1; mask < 16; mask <<= 1) p += __shfl_xor(p, mask, 32);
            part[r] = p;
        }
        if (m16 == 0) {  // lanes 0 and 16 write rows 0..7 / 8..15
#pragma unroll
            for (int r = 0; r < 8; ++r)
                scores_s[(r + rhalf) * 16 + wave] = part[r] * (1.0f / 64.0f);
        }
        __syncthreads();

        // softmax over the 16 heads, one row per lane of wave 0
        if (wave == 0 && lane < 16) {
            const int row = lane;
            float mx = scores_s[row * 16 + 0];
#pragma unroll
            for (int hh = 1; hh < 16; ++hh) mx = fmaxf(mx, scores_s[row * 16 + hh]);
            float sum = 0.0f;
            float ex[16];
#pragma unroll
            for (int hh = 0; hh < 16; ++hh) {
                ex[hh] = __expf(scores_s[row * 16 + hh] - mx);
                sum += ex[hh];
            }
            const float inv = 1.0f / sum;
#pragma unroll
            for (int hh = 0; hh < 16; ++hh) att_s[row * 16 + hh] = ex[hh] * inv;
        }
        __syncthreads();

        // accumulate att * xv into registers
#pragma unroll
        for (int r = 0; r < 8; ++r) {
            const float aw = att_s[(r + rhalf) * 16 + wave];
#pragma unroll
            for (int j = 0; j < 4; ++j) eacc[j][r] += aw * cv[j][r];
        }
        __syncthreads();  // scores_s/att_s reused next t
    }

    // combine per-head accumulators in LDS (ds_add_f32)
#pragma unroll
    for (int j = 0; j < 4; ++j) {
#pragma unroll
        for (int r = 0; r < 8; ++r) {
            atomicAdd(&emb_s[(r + rhalf) * 64 + 16 * j + m16], eacc[j][r]);
        }
    }
    __syncthreads();

    // row L2 norm of emb
    if (threadIdx.x < 16) {
        const int row = threadIdx.x;
        float ss = 0.0f;
#pragma unroll
        for (int e = 0; e < 64; ++e) {
            const float v = emb_s[row * 64 + e];
            ss += v * v;
        }
        invn_s[row] = 1.0f / fmaxf(sqrtf(ss), 1e-12f);
    }
    __syncthreads();

    // logits[16x100] = emb_n[16x64] @ Wp[64x100] + bp, WMMA over K=64.
    // Waves 0..6 cover 7 column tiles; Wp columns clamped (no divergence),
    // stores guarded at 100.
    if (wave < 7) {
        v8f cl = vzero8();
        const int n  = 16 * wave + m16;
        const int nc = min(n, 99);
        const float* __restrict__ wpp = Wp + koff * 100 + nc;
        const float* __restrict__ ep  = emb_s + m16 * 64 + koff;
        const float sc = invn_s[m16];
#pragma unroll
        for (int kk = 0; kk < 64; kk += 4) {
            v2f a; a.x = ep[kk] * sc; a.y = ep[kk + 1] * sc;
            v2f b; b.x = wpp[0]; b.y = wpp[100];
            cl = wmma4(a, b, cl);
            wpp += 400;
        }
        if (n < 100) {
            const float bb = bp[n];
#pragma unroll
            for (int r = 0; r < 8; ++r) {
                out[(size_t)(rowbase + r + rhalf) * 100 + n] = cl[r] + bb;
            }
        }
    }
}

// =====================================================================
extern "C" void kernel_launch(void* const* d_in, const int* in_sizes, int n_in,
                              void* d_out, int out_size, void* d_ws, size_t ws_size,
                              hipStream_t stream) {
    const float* feat  = (const float*)d_in[0];  // [2048,2048]
    const float* Warch = (const float*)d_in[1];  // [2048,1000]
    const float* barch = (const float*)d_in[2];  // [1000]
    const float* Q     = (const float*)d_in[3];  // [20,16,1000,64]
    const float* K     = (const float*)d_in[4];
    const float* V     = (const float*)d_in[5];
    const float* Wp    = (const float*)d_in[6];  // [64,100]
    const float* bp    = (const float*)d_in[7];  // [100]
    float* out = (float*)d_out;                  // [2048,100]

    float* h = (float*)d_ws;                     // [2048,1000] fp32, 8.19 MB

    dim3 g1(32, 8);
    k_gemm_arch<<<g1, 256, 0, stream>>>(feat, Warch, barch, h);
    k_rownorm<<<2048, 256, 0, stream>>>(h);
    k_attn<<<128, 512, 0, stream>>>(h, Q, K, V, Wp, bp, out);
}